// SparseGraphLearn_40175124086871
// MI455X (gfx1250) — compile-verified
//
#include <hip/hip_runtime.h>

typedef __attribute__((ext_vector_type(2))) float v2f;
typedef __attribute__((ext_vector_type(4))) float v4f;
typedef __attribute__((ext_vector_type(8))) float v8f;

#define N_NODES 8192
#define E_EDGES 262144
#define DIN     256
#define DOUT    128
#define ALPHA_C 0.5f

// ---------------------------------------------------------------------------
// K1: h = inputs @ weights  via v_wmma_f32_16x16x4_f32 (fp32-exact matrix op)
// One wave per 16x16 output tile. 512 M-tiles x 8 N-tiles = 4096 tiles.
// ---------------------------------------------------------------------------
__global__ __launch_bounds__(256) void gemm_h_kernel(const float* __restrict__ A,
                                                     const float* __restrict__ W,
                                                     float* __restrict__ H) {
  const int lane = threadIdx.x & 31;
  const int wave = threadIdx.x >> 5;
  const int tile = blockIdx.x * 8 + wave;   // 0..4095
  const int tm   = tile >> 3;               // 0..511
  const int tn   = tile & 7;                // 0..7
  const int half = lane >> 4;               // 0 | 1
  const int ml   = lane & 15;
  const int koff = half * 2;
  const long arow = (long)(tm * 16 + ml) * DIN + koff;
  const int  bcol = tn * 16 + ml;

  v8f c = {};
#pragma unroll 4
  for (int k0 = 0; k0 < DIN; k0 += 4) {
    v2f a = *(const v2f*)&A[arow + k0];
    v2f b;
    b.x = W[(k0 + koff + 0) * DOUT + bcol];
    b.y = W[(k0 + koff + 1) * DOUT + bcol];
    c = __builtin_amdgcn_wmma_f32_16x16x4_f32(false, a, false, b,
                                              (short)0, c, false, false);
  }
  const int ncol = tn * 16 + ml;
#pragma unroll
  for (int v = 0; v < 8; ++v) {
    const int mrow = tm * 16 + v + half * 8;
    H[(long)mrow * DOUT + ncol] = c[v];
  }
}

// ============================ CSR fast path =================================

__global__ __launch_bounds__(256) void zero_counts_kernel(int* __restrict__ cnt) {
  cnt[blockIdx.x * 256 + threadIdx.x] = 0;
}

__global__ __launch_bounds__(256) void count_kernel(const int* __restrict__ edge,
                                                    int* __restrict__ cnt) {
  const int e = blockIdx.x * 256 + threadIdx.x;
  atomicAdd(&cnt[edge[e]], 1);
}

// Exclusive scan of 8192 row counts with a single 256-thread block.
__global__ __launch_bounds__(256) void scan_kernel(const int* __restrict__ cnt,
                                                   int* __restrict__ rowptr,
                                                   int* __restrict__ cursor) {
  __shared__ int chunk[256];
  const int t = threadIdx.x;
  const int base = t * 32;
  int local[32];
  int run = 0;
#pragma unroll
  for (int k = 0; k < 32; ++k) { local[k] = run; run += cnt[base + k]; }
  chunk[t] = run;
  __syncthreads();
  for (int off = 1; off < 256; off <<= 1) {
    int v = 0;
    if (t >= off) v = chunk[t - off];
    __syncthreads();
    if (t >= off) chunk[t] += v;
    __syncthreads();
  }
  const int off0 = chunk[t] - run;   // exclusive chunk offset
#pragma unroll
  for (int k = 0; k < 32; ++k) {
    const int p = off0 + local[k];
    rowptr[base + k] = p;
    cursor[base + k] = p;
  }
}

// Per-edge score relu(|h_src - h_dst| . a); one wave per edge; append to CSR.
__global__ __launch_bounds__(256) void edge_fill_kernel(const float* __restrict__ H,
                                                        const float* __restrict__ avec,
                                                        const int* __restrict__ edge,
                                                        int* __restrict__ cursor,
                                                        int* __restrict__ colv,
                                                        float* __restrict__ valv) {
  const int lane = threadIdx.x & 31;
  const int wave = threadIdx.x >> 5;
  const int e    = blockIdx.x * 8 + wave;
  const int src  = edge[e];
  const int dst  = edge[E_EDGES + e];

  const v4f hs = *(const v4f*)&H[(long)src * DOUT + lane * 4];
  const v4f hd = *(const v4f*)&H[(long)dst * DOUT + lane * 4];
  const v4f av = *(const v4f*)&avec[lane * 4];

  float p = fabsf(hs.x - hd.x) * av.x + fabsf(hs.y - hd.y) * av.y +
            fabsf(hs.z - hd.z) * av.z + fabsf(hs.w - hd.w) * av.w;
#pragma unroll
  for (int off = 16; off > 0; off >>= 1) p += __shfl_xor(p, off, 32);

  if (lane == 0) {
    const int pos = atomicAdd(&cursor[src], 1);
    colv[pos] = dst;
    valv[pos] = fmaxf(p, 0.0f);
  }
}

// One workgroup per row: dense row in LDS (duplicates sum exactly),
// softmax locally, single NT streaming write. sgraph never re-read.
__global__ __launch_bounds__(256) void row_softmax_kernel(const int* __restrict__ rowptr,
                                                          const int* __restrict__ cnt,
                                                          const int* __restrict__ colv,
                                                          const float* __restrict__ valv,
                                                          float* __restrict__ SG) {
  __shared__ float rowv[N_NODES];   // 32 KB dense row accumulator
  __shared__ float red[256];
  const int t   = threadIdx.x;
  const int row = blockIdx.x;

#pragma unroll
  for (int j = 0; j < 32; ++j) rowv[t + 256 * j] = 0.0f;
  __syncthreads();

  const int beg = rowptr[row];
  const int end = beg + cnt[row];
  for (int j = beg + t; j < end; j += 256)
    atomicAdd(&rowv[colv[j]], valv[j]);          // ds_add_f32
  __syncthreads();

  float m = 0.0f;                                // entries >= 0
#pragma unroll
  for (int j = 0; j < 32; ++j) m = fmaxf(m, rowv[t + 256 * j]);
  red[t] = m;
  __syncthreads();
#pragma unroll
  for (int s = 128; s > 0; s >>= 1) {
    if (t < s) red[t] = fmaxf(red[t], red[t + s]);
    __syncthreads();
  }
  m = red[0];
  __syncthreads();

  float sum = 0.0f;
#pragma unroll
  for (int j = 0; j < 32; ++j) {
    const float e = __expf(rowv[t + 256 * j] - m);
    rowv[t + 256 * j] = e;                       // reuse for write pass
    sum += e;
  }
  red[t] = sum;
  __syncthreads();
#pragma unroll
  for (int s = 128; s > 0; s >>= 1) {
    if (t < s) red[t] += red[t + s];
    __syncthreads();
  }
  const float rz = 1.0f / red[0];

  const long rowbase = (long)row * N_NODES;
#pragma unroll
  for (int j = 0; j < 8; ++j) {
    const int i4 = (t + 256 * j) * 4;
    v4f y;
    y.x = rowv[i4 + 0] * rz;
    y.y = rowv[i4 + 1] * rz;
    y.z = rowv[i4 + 2] * rz;
    y.w = rowv[i4 + 3] * rz;
    __builtin_nontemporal_store(y, (v4f*)(SG + rowbase + i4));
  }
}

// ===================== dense fallback path (small ws) =======================

__global__ __launch_bounds__(256) void zero_kernel(float* __restrict__ SG) {
  const long i4 = (long)blockIdx.x * blockDim.x + threadIdx.x;
  v4f z = {0.f, 0.f, 0.f, 0.f};
  __builtin_nontemporal_store(z, (v4f*)(SG + i4 * 4));
}

__global__ __launch_bounds__(256) void edge_kernel(const float* __restrict__ H,
                                                   const float* __restrict__ avec,
                                                   const int* __restrict__ edge,
                                                   float* __restrict__ SG) {
  const int lane = threadIdx.x & 31;
  const int wave = threadIdx.x >> 5;
  const int e    = blockIdx.x * 8 + wave;
  const int src  = edge[e];
  const int dst  = edge[E_EDGES + e];

  const v4f hs = *(const v4f*)&H[(long)src * DOUT + lane * 4];
  const v4f hd = *(const v4f*)&H[(long)dst * DOUT + lane * 4];
  const v4f av = *(const v4f*)&avec[lane * 4];

  float p = fabsf(hs.x - hd.x) * av.x + fabsf(hs.y - hd.y) * av.y +
            fabsf(hs.z - hd.z) * av.z + fabsf(hs.w - hd.w) * av.w;
#pragma unroll
  for (int off = 16; off > 0; off >>= 1) p += __shfl_xor(p, off, 32);

  if (lane == 0) {
    const float v = fmaxf(p, 0.0f);
    if (v > 0.0f)
      atomicAdd(&SG[(long)src * N_NODES + dst], v);
  }
}

__global__ __launch_bounds__(256) void softmax_kernel(float* __restrict__ SG) {
  __shared__ float red[256];
  const int  t       = threadIdx.x;
  const long rowbase = (long)blockIdx.x * N_NODES;

  v4f x[8];
  float m = 0.0f;
#pragma unroll
  for (int j = 0; j < 8; ++j) {
    x[j] = __builtin_nontemporal_load((const v4f*)(SG + rowbase + (t + 256 * j) * 4));
    m = fmaxf(m, fmaxf(fmaxf(x[j].x, x[j].y), fmaxf(x[j].z, x[j].w)));
  }
  red[t] = m;
  __syncthreads();
#pragma unroll
  for (int s = 128; s > 0; s >>= 1) {
    if (t < s) red[t] = fmaxf(red[t], red[t + s]);
    __syncthreads();
  }
  m = red[0];
  __syncthreads();

  float sum = 0.0f;
#pragma unroll
  for (int j = 0; j < 8; ++j) {
    x[j].x = __expf(x[j].x - m);
    x[j].y = __expf(x[j].y - m);
    x[j].z = __expf(x[j].z - m);
    x[j].w = __expf(x[j].w - m);
    sum += (x[j].x + x[j].y) + (x[j].z + x[j].w);
  }
  red[t] = sum;
  __syncthreads();
#pragma unroll
  for (int s = 128; s > 0; s >>= 1) {
    if (t < s) red[t] += red[t + s];
    __syncthreads();
  }
  const float rz = 1.0f / red[0];

#pragma unroll
  for (int j = 0; j < 8; ++j) {
    v4f y = x[j];
    y.x *= rz; y.y *= rz; y.z *= rz; y.w *= rz;
    __builtin_nontemporal_store(y, (v4f*)(SG + rowbase + (t + 256 * j) * 4));
  }
}

// ---------------------------------------------------------------------------
// Final sparse pass: sgraph += adj * ALPHA (each edge adds 0.5; dups sum)
// ---------------------------------------------------------------------------
__global__ __launch_bounds__(256) void adj_kernel(const int* __restrict__ edge,
                                                  float* __restrict__ SG) {
  const int e = blockIdx.x * blockDim.x + threadIdx.x;
  const int src = edge[e];
  const int dst = edge[E_EDGES + e];
  atomicAdd(&SG[(long)src * N_NODES + dst], ALPHA_C);
}

extern "C" void kernel_launch(void* const* d_in, const int* in_sizes, int n_in,
                              void* d_out, int out_size, void* d_ws, size_t ws_size,
                              hipStream_t stream) {
  const float* inputs  = (const float*)d_in[0];   // [8192, 256]
  const float* weights = (const float*)d_in[1];   // [256, 128]
  const float* avec    = (const float*)d_in[2];   // [128, 1]
  const int*   edge    = (const int*)d_in[3];     // [2, E]

  float* H  = (float*)d_out;                                  // [8192,128]
  float* SG = (float*)d_out + (long)N_NODES * DOUT;           // [8192,8192]

  gemm_h_kernel<<<512, 256, 0, stream>>>(inputs, weights, H);

  // ws layout: cnt[8192] | cursor[8192] | rowptr[8192] | col[E] | val[E]
  const size_t WS_NEEDED = (size_t)3 * N_NODES * 4 + (size_t)E_EDGES * 8;
  if (d_ws != nullptr && ws_size >= WS_NEEDED) {
    char* w = (char*)d_ws;
    int*   cnt    = (int*)(w);
    int*   cursor = (int*)(w + (size_t)N_NODES * 4);
    int*   rowptr = (int*)(w + (size_t)N_NODES * 8);
    int*   colv   = (int*)(w + (size_t)N_NODES * 12);
    float* valv   = (float*)(w + (size_t)N_NODES * 12 + (size_t)E_EDGES * 4);

    zero_counts_kernel<<<N_NODES / 256, 256, 0, stream>>>(cnt);
    count_kernel<<<E_EDGES / 256, 256, 0, stream>>>(edge, cnt);
    scan_kernel<<<1, 256, 0, stream>>>(cnt, rowptr, cursor);
    edge_fill_kernel<<<E_EDGES / 8, 256, 0, stream>>>(H, avec, edge, cursor, colv, valv);
    row_softmax_kernel<<<N_NODES, 256, 0, stream>>>(rowptr, cnt, colv, valv, SG);
  } else {
    zero_kernel<<<65536, 256, 0, stream>>>(SG);
    edge_kernel<<<E_EDGES / 8, 256, 0, stream>>>(H, avec, edge, SG);
    softmax_kernel<<<N_NODES, 256, 0, stream>>>(SG);
  }

  adj_kernel<<<E_EDGES / 256, 256, 0, stream>>>(edge, SG);
}